// SGN_55353538510974
// MI455X (gfx1250) — compile-verified
//
#include <hip/hip_runtime.h>
#include <math.h>

#define Bsz  256
#define Nn   128
#define Ff   64
#define Ee   8
#define Ss   7
#define Qq   2
#define Kk   15
#define OUTn 2
#define CIN  136        // 2F + E
#define KTOT 1224       // 9 * CIN
#define KB4  (KTOT / 4) // 306 k-blocks of 4

typedef float v2f __attribute__((ext_vector_type(2)));
typedef float v8f __attribute__((ext_vector_type(8)));

__device__ __forceinline__ float sigmoidf_(float x) { return 1.0f / (1.0f + expf(-x)); }

__device__ __forceinline__ float wredMax(float v) {
#pragma unroll
    for (int o = 16; o > 0; o >>= 1) v = fmaxf(v, __shfl_down(v, o, 32));
    return v;
}
__device__ __forceinline__ float wredMin(float v) {
#pragma unroll
    for (int o = 16; o > 0; o >>= 1) v = fminf(v, __shfl_down(v, o, 32));
    return v;
}
__device__ __forceinline__ float wredSum(float v) {
#pragma unroll
    for (int o = 16; o > 0; o >>= 1) v += __shfl_down(v, o, 32);
    return v;
}

__global__ __launch_bounds__(128) void sgn_kernel(
    const float* __restrict__ X,     const float* __restrict__ A,
    const float* __restrict__ Aattr, const float* __restrict__ noise,
    const float* __restrict__ wq0,   const float* __restrict__ wq,
    const float* __restrict__ lf,    const float* __restrict__ li,
    const float* __restrict__ lc,    const float* __restrict__ lo,
    const float* __restrict__ c0,    const float* __restrict__ c1,
    const float* __restrict__ mlp,   const float* __restrict__ att,
    float* __restrict__ out)
{
    // ---- dynamic LDS layout ----
    extern __shared__ float smem[];
    float* XS   = smem;                 // 8192 floats: Xg [128][64], reused as struct [7][7][136]
    float* Bsw  = XS + Nn * Ff;         // 19584 floats: cnn0 pre-swizzled into WMMA B lane layout
    float* zPad = Bsw + KB4 * 64;       // 144 floats: zero pad for invalid M rows
    float* hbuf = zPad + 144;           // 400 floats: conv0 output [25][16]

    __shared__ float sc[Nn];
    __shared__ float accS[Ss * Ff];
    __shared__ float cellS[Ff], hiddenS[Ff];
    __shared__ float gf[Ff], gi[Ff], gc[Ff], go[Ff];
    __shared__ float visited[Nn];
    __shared__ float feats[Qq * Kk];
    __shared__ float rs[4];
    __shared__ int   rsi[4];
    __shared__ float bc[2];
    __shared__ int   selS[Ss];

    const int b    = blockIdx.x;
    const int t    = threadIdx.x;
    const int lane = t & 31;
    const int wv   = t >> 5;

    // ---- block-wide wave32 reductions (all 128 threads must call) ----
    auto blockMax = [&](float v) -> float {
        v = wredMax(v);
        if (lane == 0) rs[wv] = v;
        __syncthreads();
        float r = fmaxf(fmaxf(rs[0], rs[1]), fmaxf(rs[2], rs[3]));
        __syncthreads();
        return r;
    };
    auto blockMin = [&](float v) -> float {
        v = wredMin(v);
        if (lane == 0) rs[wv] = v;
        __syncthreads();
        float r = fminf(fminf(rs[0], rs[1]), fminf(rs[2], rs[3]));
        __syncthreads();
        return r;
    };
    auto blockSum = [&](float v) -> float {
        v = wredSum(v);
        if (lane == 0) rs[wv] = v;
        __syncthreads();
        float r = (rs[0] + rs[1]) + (rs[2] + rs[3]);
        __syncthreads();
        return r;
    };
    auto blockArgmax = [&](float v, int i) -> int {
#pragma unroll
        for (int o = 16; o > 0; o >>= 1) {
            float v2 = __shfl_down(v, o, 32);
            int   i2 = __shfl_down(i, o, 32);
            if (v2 > v || (v2 == v && i2 < i)) { v = v2; i = i2; }
        }
        if (lane == 0) { rs[wv] = v; rsi[wv] = i; }
        __syncthreads();
        float bv = rs[0]; int bi = rsi[0];
#pragma unroll
        for (int w = 1; w < 4; ++w)
            if (rs[w] > bv || (rs[w] == bv && rsi[w] < bi)) { bv = rs[w]; bi = rsi[w]; }
        __syncthreads();
        return bi;
    };

    // ---- LSTM step: xx = [cell, Xg[node]]; 2 gate matvecs per thread half ----
    auto lstm_step = [&](int node) {
        const int j = t & 63;
        const float* W0 = (t < 64) ? lf : li;   // f / i
        const float* W1 = (t < 64) ? lc : lo;   // c / o
        float d0 = 0.f, d1 = 0.f;
        for (int k = 0; k < 2 * Ff; ++k) {
            float xv = (k < Ff) ? cellS[k] : XS[node * Ff + (k - Ff)];
            d0 += xv * W0[k * Ff + j];
            d1 += xv * W1[k * Ff + j];
        }
        if (t < 64) { gf[j] = sigmoidf_(d0); gc[j] = tanhf(d1); }
        else        { gi[j] = sigmoidf_(d0); go[j] = sigmoidf_(d1); }
        __syncthreads();
        if (t < Ff) {
            float cn = cellS[t] * gf[t] + gi[t] * gc[t];
            cellS[t]   = cn;
            hiddenS[t] = tanhf(cn) * go[t];
        }
        __syncthreads();
    };

    // ---- one-time staging: swizzle cnn0 into WMMA B layout; zero the pad ----
    // B operand for k-block kb4: lane l supplies K = kb4*4 + (l>=16 ? 2:0) + u, N = l%16
    for (int idx = t; idx < KB4 * 64; idx += 128) {
        int kb4 = idx >> 6;
        int l   = (idx >> 1) & 31;
        int u   = idx & 1;
        int k   = kb4 * 4 + ((l >> 4) << 1) + u;
        int col = l & 15;
        Bsw[idx] = (col < Kk) ? c0[k * Kk + col] : 0.f;
    }
    for (int idx = t; idx < 144; idx += 128) zPad[idx] = 0.f;
    __syncthreads();

    for (int q = 0; q < Qq; ++q) {
        // ---- (re)load Xg, reset per-query state ----
        for (int idx = t; idx < Nn * Ff; idx += 128) XS[idx] = X[(size_t)b * Nn * Ff + idx];
        visited[t] = 0.f;
        if (t < Ff) { cellS[t] = 0.f; hiddenS[t] = 0.f; }
        __syncthreads();

        // ================= step 0 =================
        int cur;
        {
            const float* w = wq0 + q * Ff;
            float s = 0.f;
            for (int f = 0; f < Ff; ++f) s += XS[t * Ff + f] * w[f];
            float mn = blockMin(s);
            float mx = blockMax(s);
            float sn = (s - mn) / (mx - mn + 1e-9f);
            sc[t] = sn;
            __syncthreads();
            int am = blockArgmax(sn + noise[(((size_t)b * Qq + q) * Ss + 0) * Nn + t], t);
            float smx  = blockMax(sn);
            float esum = blockSum(expf(sn - smx));
            float asel = expf(sc[am] - smx) / esum;
            if (t < Ff)  accS[t] = XS[am * Ff + t] * asel;
            if (t == am) visited[t] = 1.f;
            if (t == 0)  selS[0] = am;
            __syncthreads();
            lstm_step(am);
            cur = am;
        }

        // ================= steps 1..S-1 =================
        for (int step = 1; step < Ss; ++step) {
            const float* w = wq + (size_t)(q * (Ss - 1) + (step - 1)) * CIN;
            float s = 0.f;
            for (int k = 0; k < Ff; ++k) s += hiddenS[k] * w[k];
            for (int k = 0; k < Ff; ++k) s += XS[t * Ff + k] * w[Ff + k];
            const float* ea = Aattr + (((size_t)b * Nn + cur) * Nn + t) * Ee;
#pragma unroll
            for (int e = 0; e < Ee; ++e) s += ea[e] * w[2 * Ff + e];
            float mn = blockMin(s);
            float mx = blockMax(s);
            float sn = (s - mn) / (mx - mn + 1e-9f);
            float cand = A[((size_t)b * Nn + t) * Nn + cur] * (1.f - visited[t]);
            sn = (sn + 1000.f) * cand - 1000.f;
            sc[t] = sn;
            __syncthreads();
            int am = blockArgmax(sn + noise[(((size_t)b * Qq + q) * Ss + step) * Nn + t], t);
            float smx  = blockMax(sn);
            float esum = blockSum(expf(sn - smx));
            float asel = expf(sc[am] - smx) / esum;
            if (t < Ff)  accS[step * Ff + t] = XS[am * Ff + t] * asel;
            if (t == am) visited[t] = 1.f;
            if (t == 0)  selS[step] = am;
            __syncthreads();
            lstm_step(am);
            cur = am;
        }

        // ---- build struct [7][7][136] in-place (Xg no longer needed) ----
        for (int idx = t; idx < Ss * Ss * CIN; idx += 128) {
            int c  = idx % CIN;
            int ij = idx / CIN;
            int i  = ij / Ss, jn = ij % Ss;
            float v;
            if (c < Ff)          v = accS[i * Ff + c];
            else if (c < 2 * Ff) v = accS[jn * Ff + (c - Ff)];
            else {
                int si = selS[i], sj = selS[jn];
                float a = A[((size_t)b * Nn + si) * Nn + sj];
                v = Aattr[(((size_t)b * Nn + si) * Nn + sj) * Ee + (c - 2 * Ff)] * a;
            }
            XS[idx] = v;
        }
        __syncthreads();

        // ---- conv0 implicit GEMM via V_WMMA_F32_16X16X4_F32 ----
        // M = 25 positions (2 tiles of 16 on waves 0,1), N = 15 ch (pad 16), K = 1224
        // Inner loop: two 8B-aligned ds_load_b64 + wmma, additive addressing, no EXEC churn.
        if (wv < 2) {
            const int  half = lane >> 4;
            const int  lm   = lane & 15;
            const int  M    = wv * 16 + lm;
            const int  oy   = M / 5, ox = M - oy * 5;
            const bool mok  = (M < 25);
            v8f acc8 = { 0.f, 0.f, 0.f, 0.f, 0.f, 0.f, 0.f, 0.f };
            const float* bPtr = Bsw + lane * 2;
#pragma unroll
            for (int kw = 0; kw < 9; ++kw) {
                const int ky = kw / 3, kx = kw - ky * 3;
                const float* aPtr =
                    (mok ? (XS + ((oy + ky) * Ss + (ox + kx)) * CIN) : zPad) + half * 2;
#pragma unroll 2
                for (int cc = 0; cc < 34; ++cc) {
                    v2f a  = *(const v2f*)aPtr;
                    v2f bb = *(const v2f*)bPtr;
                    acc8 = __builtin_amdgcn_wmma_f32_16x16x4_f32(
                        /*neg_a=*/false, a, /*neg_b=*/false, bb,
                        /*c_mod=*/(short)0, acc8, /*reuse_a=*/false, /*reuse_b=*/false);
                    aPtr += 4;
                    bPtr += 64;
                }
            }
#pragma unroll
            for (int v = 0; v < 8; ++v) {
                int row = wv * 16 + half * 8 + v;
                int col = lm;
                if (row < 25 && col < Kk) hbuf[row * 16 + col] = fmaxf(acc8[v], 0.f);
            }
        }
        __syncthreads();

        // ---- conv1 (5x5x15 -> 15), gate, feats ----
        if (t < Kk) {
            float s = 0.f;
            for (int p = 0; p < 25; ++p)
                for (int ci = 0; ci < Kk; ++ci)
                    s += hbuf[p * 16 + ci] * c1[((size_t)(p * Kk + ci)) * Kk + t];
            sc[t] = fmaxf(s, 0.f);
        }
        __syncthreads();
        if (t == 0) {
            float g = 0.f;
            for (int ci = 0; ci < Kk; ++ci) g += sc[ci] * att[ci];
            bc[0] = sigmoidf_(g);
        }
        __syncthreads();
        if (t < Kk) feats[q * Kk + t] = sc[t] * bc[0];
        __syncthreads();
    }

    // ---- final MLP: [Q*K=30] @ [30,2] ----
    if (t < OUTn) {
        float s = 0.f;
        for (int jn = 0; jn < Qq * Kk; ++jn) s += feats[jn] * mlp[jn * OUTn + t];
        out[b * OUTn + t] = s;
    }
}

extern "C" void kernel_launch(void* const* d_in, const int* in_sizes, int n_in,
                              void* d_out, int out_size, void* d_ws, size_t ws_size,
                              hipStream_t stream) {
    (void)in_sizes; (void)n_in; (void)out_size; (void)d_ws; (void)ws_size;
    const size_t shmem = (size_t)(Nn * Ff + KB4 * 64 + 144 + 400) * sizeof(float); // ~111 KB
    sgn_kernel<<<Bsz, 128, shmem, stream>>>(
        (const float*)d_in[0],  (const float*)d_in[1],  (const float*)d_in[2],
        (const float*)d_in[3],  (const float*)d_in[4],  (const float*)d_in[5],
        (const float*)d_in[6],  (const float*)d_in[7],  (const float*)d_in[8],
        (const float*)d_in[9],  (const float*)d_in[10], (const float*)d_in[11],
        (const float*)d_in[12], (const float*)d_in[13], (float*)d_out);
}